// MultiHeadedAttention_3_18631568130106
// MI455X (gfx1250) — compile-verified
//
#include <hip/hip_runtime.h>
#include <hip/hip_bf16.h>
#include <math.h>

typedef __attribute__((ext_vector_type(2))) float v2f;
typedef __attribute__((ext_vector_type(8))) float v8f;

#define DCH   256      // channel dim of query/key
#define HEADS 8
#define LENC  516      // 2*D + 4
#define NCOL  256      // N (dim-2 length)

// LDS layout for W^T (zero-padded to 16 heads), arranged so each lane's WMMA
// B-fragment pair is CONTIGUOUS (one ds_load_b64, no repacking):
// per K-group g (channels 4g..4g+3), 128 floats:
//   [g*128 +  0 + 2n+w] = W[n][4g      + w]   (query, lanes 0-15,  w=0,1)
//   [g*128 + 32 + 2n+w] = W[n][4g + 2  + w]   (query, lanes 16-31)
//   [g*128 + 64 + 2n+w] = W[n][256+4g  + w]   (key,   lanes 0-15)
//   [g*128 + 96 + 2n+w] = W[n][256+4g+2+ w]   (key,   lanes 16-31)
// spat block at 8192 (64 floats):
//   [8192 + s*32 + 2n+w] = W[n][512 + 2s + w] (s=0: s_query rows, s=1: s_key)
#define SW_ELEMS (64 * 128 + 64)   // 8256 floats = 33 KB

// ---------------------------------------------------------------------------
// Kernel 1: fused linear (2x WMMA f32 16x16x4 per K-step) + cosine scores.
// One wave = one 16-pixel M-tile; N-tile = 16 (8 real heads + 8 zero pad).
// Each q/k element is loaded from global exactly once (non-temporal).
// ---------------------------------------------------------------------------
__global__ __launch_bounds__(256)
void fused_linear_cos_kernel(const float* __restrict__ query,
                             const float* __restrict__ key,
                             const float* __restrict__ s_query,
                             const float* __restrict__ s_key,
                             const float* __restrict__ roi,
                             const float* __restrict__ W,
                             const float* __restrict__ b_lin,
                             float* __restrict__ out,
                             float* __restrict__ scores)
{
    __shared__ float sW[SW_ELEMS];
    for (int idx = threadIdx.x; idx < SW_ELEMS; idx += blockDim.x) {
        int n, c;
        if (idx < 8192) {
            int g     = idx >> 7;          // K-group
            int local = idx & 127;
            int sub   = local >> 5;        // 0:q/lo 1:q/hi 2:k/lo 3:k/hi
            n         = (local >> 1) & 15; // head (padded to 16)
            int w     = local & 1;
            c = ((sub >> 1) ? DCH : 0) + 4 * g + ((sub & 1) << 1) + w;
        } else {
            int local = idx - 8192;
            int sub   = local >> 5;        // 0: s_query rows, 1: s_key rows
            n         = (local >> 1) & 15;
            int w     = local & 1;
            c = 512 + (sub << 1) + w;
        }
        sW[idx] = (n < HEADS) ? W[n * LENC + c] : 0.0f;
    }
    __syncthreads();

    const int  lane = threadIdx.x & 31;
    const int  wave = threadIdx.x >> 5;
    const int  m    = lane & 15;          // pixel-in-tile (A) / head col (B,D)
    const int  hi   = lane >> 4;          // half-wave select
    const int  koff = hi << 1;            // K sub-offset: 0 or 2
    const long tilebase = (long)blockIdx.x * 128 + (long)wave * 16;
    const long p = tilebase + m;          // flat pixel (b*N + i)*N + j

    const float* qrow = query + p * DCH + koff;
    const float* krow = key   + p * DCH + koff;
    const float* wp   = sW + hi * 32 + 2 * m;   // per-lane B-fragment base

    v8f acc  = {0.f,0.f,0.f,0.f,0.f,0.f,0.f,0.f};
    v2f qkv  = {0.f, 0.f};
    v2f qqv  = {0.f, 0.f};
    v2f kkv  = {0.f, 0.f};

    // one-step software pipeline over the merged q/k K-loop (64 steps)
    v2f aq = __builtin_nontemporal_load((const v2f*)qrow);
    v2f ak = __builtin_nontemporal_load((const v2f*)krow);

    for (int k0 = 0; k0 < DCH; k0 += 4) {
        v2f aq_n, ak_n;
        if (k0 + 4 < DCH) {   // prefetch next step; skipped on the last step
            aq_n = __builtin_nontemporal_load((const v2f*)(qrow + k0 + 4));
            ak_n = __builtin_nontemporal_load((const v2f*)(krow + k0 + 4));
        }
        // cosine partial sums, packed (v_pk_fma_f32 on the loaded pairs)
        qkv += aq * ak;
        qqv += aq * aq;
        kkv += ak * ak;

        // B fragments: contiguous per-lane pairs (ds_load_b64 / 2addr_b64)
        v2f bq = *(const v2f*)(wp + k0 * 32);        // query rows for this step
        v2f bk = *(const v2f*)(wp + k0 * 32 + 64);   // key rows (+256 B)
        acc = __builtin_amdgcn_wmma_f32_16x16x4_f32(false, aq, false, bq,
                                                    (short)0, acc, false, false);
        acc = __builtin_amdgcn_wmma_f32_16x16x4_f32(false, ak, false, bk,
                                                    (short)0, acc, false, false);
        aq = aq_n;
        ak = ak_n;
    }

    // ---- channels [512,516) = concat(s_query, s_key) ----
    {
        const float* sp = hi ? (s_key + p * 2) : (s_query + p * 2);
        v2f a = *(const v2f*)sp;
        v2f b = *(const v2f*)(sW + 8192 + hi * 32 + 2 * m);
        acc = __builtin_amdgcn_wmma_f32_16x16x4_f32(false, a, false, b,
                                                    (short)0, acc, false, false);
    }

    // ---- cosine score: combine halves + half-wave partials (xor-16) ----
    float qk = qkv.x + qkv.y;
    float qq = qqv.x + qqv.y;
    float kk = kkv.x + kkv.y;
    qk += __shfl_xor(qk, 16, 32);
    qq += __shfl_xor(qq, 16, 32);
    kk += __shfl_xor(kk, 16, 32);
    if (hi == 0) {
        const float eps = 1e-8f;
        float s = qk / (fmaxf(sqrtf(qq), eps) * fmaxf(sqrtf(kk), eps));
        scores[p] = fmaxf(s, 0.0f);
    }

    // ---- epilogue: D layout = lane n=m, row M = g + 8*hi; sigmoid + roi ----
    const int   head = m;
    const float bias = (head < HEADS) ? b_lin[head] : 0.0f;
    #pragma unroll
    for (int g = 0; g < 8; ++g) {
        long pg = tilebase + g + (hi << 3);
        if (head < HEADS) {
            float v = acc[g] + bias;
            v = 1.0f / (1.0f + __expf(-v));
            out[pg * HEADS + head] = v * roi[pg];
        }
    }
}

// ---------------------------------------------------------------------------
// Kernel 2: zero the indicator vector (re-init every call; ws not re-poisoned)
// ---------------------------------------------------------------------------
__global__ void zero_ind_kernel(float* __restrict__ ind)
{
    ind[threadIdx.x] = 0.0f;
}

// ---------------------------------------------------------------------------
// Kernel 3: per-row top-k membership via rank counting (ties -> lower index),
// union into global indicator. Concurrent stores of 1.0f are benign.
// ---------------------------------------------------------------------------
__global__ __launch_bounds__(NCOL)
void topk_union_kernel(const float* __restrict__ scores,
                       const int* __restrict__ node_num_p,
                       float* __restrict__ ind)
{
    __shared__ float s[NCOL];
    const int t = threadIdx.x;
    const int r = blockIdx.x;
    const float v = scores[(long)r * NCOL + t];
    s[t] = v;
    __syncthreads();
    int cnt = 0;
    #pragma unroll 8
    for (int j = 0; j < NCOL; ++j) {
        float u = s[j];
        cnt += (u > v) || (u == v && j < t);
    }
    if (cnt < node_num_p[0]) ind[t] = 1.0f;
}

// ---------------------------------------------------------------------------
// Kernel 4: out[b,i,j,h] *= ind[j]
// ---------------------------------------------------------------------------
__global__ void apply_ind_kernel(float* __restrict__ out,
                                 const float* __restrict__ ind, int n)
{
    int idx = blockIdx.x * blockDim.x + threadIdx.x;
    if (idx < n) {
        int j = (idx >> 3) & (NCOL - 1);   // heads = 8
        out[idx] *= ind[j];
    }
}

extern "C" void kernel_launch(void* const* d_in, const int* in_sizes, int n_in,
                              void* d_out, int out_size, void* d_ws, size_t ws_size,
                              hipStream_t stream)
{
    const float* query   = (const float*)d_in[0];
    const float* key     = (const float*)d_in[1];
    const float* s_query = (const float*)d_in[2];
    const float* s_key   = (const float*)d_in[3];
    const float* roi     = (const float*)d_in[4];
    const float* W       = (const float*)d_in[5];
    const float* b_lin   = (const float*)d_in[6];
    const int*   node    = (const int*)d_in[7];

    float* out    = (float*)d_out;
    float* scores = (float*)d_ws;            // P floats
    const int P   = in_sizes[4];             // B*N*N = 131072
    float* ind    = scores + P;              // NCOL floats

    // 1) fused WMMA linear + cosine scores (128 pixels / block)
    fused_linear_cos_kernel<<<P / 128, 256, 0, stream>>>(
        query, key, s_query, s_key, roi, W, b_lin, out, scores);

    // 2) reset indicator
    zero_ind_kernel<<<1, NCOL, 0, stream>>>(ind);

    // 3) top-k union over all B*N rows
    topk_union_kernel<<<P / NCOL, NCOL, 0, stream>>>(scores, node, ind);

    // 4) apply indicator along dim 2
    apply_ind_kernel<<<(out_size + 255) / 256, 256, 0, stream>>>(out, ind, out_size);
}